// PEPS_15934328668968
// MI455X (gfx1250) — compile-verified
//
#include <hip/hip_runtime.h>
#include <math.h>

// PEPS amplitude via boundary-MPS zip-up, MI455X (gfx1250, wave32).
// One workgroup (256 threads = 8 wave32) per sample. Dense contractions use
// V_WMMA_F32_16X16X4_F32 on LDS-resident tiles. Waves compute 32x32 output
// blocks (2x2 WMMA tiles) so each K-step issues 4 independent wmmas off 8
// fragment loads (halves LDS bytes/wmma vs 1x1 tiling). SVD truncation is
// replaced by an orthogonal-subspace truncation built from WMMA GEMMs.
// Global->LDS staging uses async-to-LDS ops (ASYNCcnt).

#define RGRID 10
#define CGRID 10
#define DB    6
#define CHI   36
#define KQ    36
#define MAXR  216
#define MAXC  216
#define THREADS 256
#define NWAVES  8
#define SITE_F  7776            // 36*6*36
#define SAMPLE_F (CGRID * SITE_F)

typedef __attribute__((ext_vector_type(2))) float v2f;
typedef __attribute__((ext_vector_type(8))) float v8f;

struct TensPtrs { const float* t[RGRID * CGRID]; };

#define WMMA4(AF, BF, ACC)                                                    \
  __builtin_amdgcn_wmma_f32_16x16x4_f32(false, (AF), false, (BF), (short)0,   \
                                        (ACC), false, false)

// Branchless guarded LDS read: unconditional ds_load + cndmask (no saveexec).
__device__ __forceinline__ float ldsel(const float* p, int idx, bool ok) {
  float v = p[ok ? idx : 0];
  return ok ? v : 0.f;
}

// Async global -> LDS copy (per-lane gather), tracked by ASYNCcnt.
__device__ __forceinline__ void async_g2l(float* dst, const float* src, int n) {
  for (int e = threadIdx.x; e < n; e += THREADS) {
    unsigned lds = (unsigned)(uintptr_t)(dst + e);
    const float* g = src + e;
    asm volatile("global_load_async_to_lds_b32 %0, %1, off"
                 : : "v"(lds), "v"(g) : "memory");
  }
  asm volatile("s_wait_asynccnt 0x0" : : : "memory");
}

// Guarded 16x16 tile (edge path): branchless cndmask loads, guarded stores.
__device__ __forceinline__ void tile16_guarded(float* C, int cLD,
                                               const float* A, int aRS, int aCS,
                                               const float* B, int bRS, int bCS,
                                               int M, int N, int K,
                                               int mm, int nn,
                                               int l16, int h2, int half) {
  const int ar = mm + l16;
  const int bc = nn + l16;
  const bool aok = ar < M, bok = bc < N;
  v8f acc = {0.f, 0.f, 0.f, 0.f, 0.f, 0.f, 0.f, 0.f};
  for (int k0 = 0; k0 < K; k0 += 4) {
    const int ka = k0 + h2;
    v2f af, bf;
    af.x = ldsel(A, ar * aRS + ka * aCS, aok && ka < K);
    af.y = ldsel(A, ar * aRS + (ka + 1) * aCS, aok && ka + 1 < K);
    bf.x = ldsel(B, ka * bRS + bc * bCS, bok && ka < K);
    bf.y = ldsel(B, (ka + 1) * bRS + bc * bCS, bok && ka + 1 < K);
    acc = WMMA4(af, bf, acc);
  }
#pragma unroll
  for (int i = 0; i < 8; ++i) {
    const int rr = mm + i + 8 * half;
    if (rr < M && bc < N) C[rr * cLD + bc] = acc[i];
  }
}

// C(M,N) = A(M,K) @ B(K,N); element (r,c) of A at A[r*aRS + c*aCS].
// 8 waves split 32x32 output blocks; fully in-bounds blocks run the 2x2
// register-tiled fast path (4 back-to-back wmmas per K-step); ragged blocks
// fall back to guarded 16x16 tiles.
__device__ __forceinline__ void gemm_f32(float* C, int cLD,
                                         const float* A, int aRS, int aCS,
                                         const float* B, int bRS, int bCS,
                                         int M, int N, int K) {
  const int tid  = threadIdx.x;
  const int wave = tid >> 5;
  const int lane = tid & 31;
  const int half = lane >> 4;     // lanes 0-15 vs 16-31
  const int l16  = lane & 15;
  const int h2   = 2 * half;      // VGPR0 holds K=+0/+2, VGPR1 K=+1/+3
  const int mT2 = (M + 31) >> 5;
  const int nT2 = (N + 31) >> 5;
  const int K4 = K & ~3;
  for (int t = wave; t < mT2 * nT2; t += NWAVES) {
    const int tm = t / nT2;
    const int tn = t - tm * nT2;
    const int m0 = tm << 5;
    const int n0 = tn << 5;
    if (m0 + 32 <= M && n0 + 32 <= N) {
      const int ar0 = m0 + l16;
      const int bc0 = n0 + l16;
      const float* Ap0 = A + ar0 * aRS;
      const float* Ap1 = Ap0 + 16 * aRS;
      const float* Bp0 = B + bc0 * bCS;
      const float* Bp1 = Bp0 + 16 * bCS;
      v8f a00 = {0.f, 0.f, 0.f, 0.f, 0.f, 0.f, 0.f, 0.f};
      v8f a01 = a00, a10 = a00, a11 = a00;
#pragma unroll 2
      for (int k0 = 0; k0 < K4; k0 += 4) {
        const int i0 = (k0 + h2) * aCS, i1 = (k0 + h2 + 1) * aCS;
        const int j0 = (k0 + h2) * bRS, j1 = (k0 + h2 + 1) * bRS;
        v2f af0, af1, bf0, bf1;
        af0.x = Ap0[i0]; af0.y = Ap0[i1];
        af1.x = Ap1[i0]; af1.y = Ap1[i1];
        bf0.x = Bp0[j0]; bf0.y = Bp0[j1];
        bf1.x = Bp1[j0]; bf1.y = Bp1[j1];
        a00 = WMMA4(af0, bf0, a00);
        a01 = WMMA4(af0, bf1, a01);
        a10 = WMMA4(af1, bf0, a10);
        a11 = WMMA4(af1, bf1, a11);
      }
      if (K4 < K) {               // K remainder (K in {1,6,...})
        const int ka = K4 + h2;
        const bool o0 = ka < K, o1 = ka + 1 < K;
        v2f af0, af1, bf0, bf1;
        af0.x = ldsel(Ap0, ka * aCS, o0);
        af0.y = ldsel(Ap0, (ka + 1) * aCS, o1);
        af1.x = ldsel(Ap1, ka * aCS, o0);
        af1.y = ldsel(Ap1, (ka + 1) * aCS, o1);
        bf0.x = ldsel(Bp0, ka * bRS, o0);
        bf0.y = ldsel(Bp0, (ka + 1) * bRS, o1);
        bf1.x = ldsel(Bp1, ka * bRS, o0);
        bf1.y = ldsel(Bp1, (ka + 1) * bRS, o1);
        a00 = WMMA4(af0, bf0, a00);
        a01 = WMMA4(af0, bf1, a01);
        a10 = WMMA4(af1, bf0, a10);
        a11 = WMMA4(af1, bf1, a11);
      }
#pragma unroll
      for (int i = 0; i < 8; ++i) {
        const int r0 = m0 + i + 8 * half;
        C[r0 * cLD + bc0]        = a00[i];
        C[r0 * cLD + bc0 + 16]   = a01[i];
        C[(r0 + 16) * cLD + bc0]      = a10[i];
        C[(r0 + 16) * cLD + bc0 + 16] = a11[i];
      }
    } else {
#pragma unroll
      for (int sm = 0; sm < 2; ++sm) {
#pragma unroll
        for (int sn = 0; sn < 2; ++sn) {
          const int mm = m0 + (sm << 4);
          const int nn = n0 + (sn << 4);
          if (mm < M && nn < N)
            tile16_guarded(C, cLD, A, aRS, aCS, B, bRS, bCS, M, N, K,
                           mm, nn, l16, h2, half);
        }
      }
    }
  }
}

// One classical Gram-Schmidt sweep using G = Q^T Q (WMMA) + column norm.
// Q: rows x kq, leading dim KQ. Qn scratch >= rows*KQ. G >= KQ*KQ.
__device__ void gs_orth(float* Q, float* Qn, float* G, int rows, int kq) {
  __syncthreads();
  gemm_f32(G, KQ, Q, 1, KQ, Q, KQ, 1, kq, kq, rows);   // G = Q^T @ Q
  __syncthreads();
  for (int e = threadIdx.x; e < rows * KQ; e += THREADS) {
    const int j = e % KQ, row = e / KQ;                // compile-time /36
    if (j < kq) {
      float acc = Q[e];
      for (int jp = 0; jp < j; ++jp) {
        float dia = G[jp * KQ + jp];
        dia = dia > 1e-20f ? dia : 1e-20f;
        acc -= (G[jp * KQ + j] / dia) * Q[row * KQ + jp];
      }
      Qn[e] = acc;
    }
  }
  __syncthreads();
  if ((int)threadIdx.x < kq) {
    const int j = threadIdx.x;
    float n2 = 0.f;
    for (int rr = 0; rr < rows; ++rr) { float q = Qn[rr * KQ + j]; n2 += q * q; }
    G[j] = rsqrtf(n2 > 1e-30f ? n2 : 1e-30f);
  }
  __syncthreads();
  for (int e = threadIdx.x; e < rows * KQ; e += THREADS) {
    const int j = e % KQ;
    Q[e] = (j < kq) ? Qn[e] * G[j] : 0.f;
  }
  __syncthreads();
}

__global__ __launch_bounds__(THREADS)
void peps_zipup_kernel(TensPtrs tp, const int* __restrict__ x,
                       float* __restrict__ out, float* __restrict__ ws) {
  extern __shared__ char smemRaw[];
  float* mat = (float*)smemRaw;        // 46656 f
  float* mB  = mat + MAXR * MAXC;      // boundary site     7776 f
  float* Qb  = mB  + SITE_F;           // Q panel           7776 f
  float* Zb  = Qb  + MAXR * KQ;        // Z panel / carry   7776 f
  float* Tb  = Zb  + MAXR * KQ;        // cm slice / GS tmp 7776 f
  float* wB  = Tb  + MAXR * KQ;        // MPO site          1296 f
  float* Gb  = wB  + 1296;             // Gram              1296 f
  float* vA  = Gb  + 1296;             // final vec A         40 f
  float* vB2 = vA  + 40;               // final vec B         40 f
  int*   dI  = (int*)(vB2 + 40);       // Dl[10], Dr[10]

  const int s   = blockIdx.x;
  const int tid = threadIdx.x;
  float* wss = ws + (size_t)s * SAMPLE_F;

  if (tid < CGRID) {                   // boundary MPS init: (1,1,1) = 1.0
    wss[tid * SITE_F] = 1.0f;
    dI[tid] = 1;
    dI[CGRID + tid] = 1;
  }
  __syncthreads();

#pragma unroll 1
  for (int r = 0; r < RGRID; ++r) {
    const int uN = (r == 0)         ? 1 : DB;
    const int dN = (r == RGRID - 1) ? 1 : DB;
    int kIn = 1;
    if (tid == 0) Zb[0] = 1.0f;        // carry = 1x1 identity
    __syncthreads();

#pragma unroll 1
    for (int c = 0; c < CGRID; ++c) {
      const int wlN = (c == 0)         ? 1 : DB;
      const int wrN = (c == CGRID - 1) ? 1 : DB;
      const int Dl  = dI[c];
      const int Dr  = dI[CGRID + c];
      const int rows = kIn * dN;
      const int cols = Dr * wrN;
      const int spin = x[s * (RGRID * CGRID) + r * CGRID + c];
      const float* trc = tp.t[r * CGRID + c];

      // Stage boundary site m (Dl,uN,Dr) via async global->LDS copy
      async_g2l(mB, wss + c * SITE_F, Dl * uN * Dr);
      // Stage MPO site w[l][rt][u][d] = t[spin][u][d][l][rt] (async gather)
      const int wCnt = wlN * wrN * uN * dN;
      for (int e = tid; e < wCnt; e += THREADS) {
        int d_ = e % dN;  int tm = e / dN;
        int u_ = tm % uN; tm /= uN;
        int rt = tm % wrN;
        int l_ = tm / wrN;
        unsigned lds = (unsigned)(uintptr_t)(wB + e);
        const float* g = trc + ((((spin * uN + u_) * dN + d_) * wlN + l_) * wrN + rt);
        asm volatile("global_load_async_to_lds_b32 %0, %1, off"
                     : : "v"(lds), "v"(g) : "memory");
      }
      asm volatile("s_wait_asynccnt 0x0" : : : "memory");
      __syncthreads();

      // mat[(k,d)][(Dr,wr)] = sum_{wl,p,Dl} carry[k,Dl,wl] m[Dl,p,Dr] w[wl,wr,p,d]
      for (int e = tid; e < rows * cols; e += THREADS) mat[e] = 0.f;

      // Column decomposition hoisted: each thread owns one output column.
      const int col = tid;
      const bool cok = col < cols;
      const int dr_ = cok ? col / wrN : 0;   // once per site, not per row
      const int wr_ = cok ? col - dr_ * wrN : 0;

#pragma unroll 1
      for (int wl = 0; wl < wlN; ++wl) {
        __syncthreads();
        // cm = carry(:, :, wl) @ m  ->  Tb[kIn x (uN*Dr)]
        gemm_f32(Tb, uN * Dr,
                 Zb + wl, Dl * wlN, wlN,    // A: carry slice (kIn x Dl)
                 mB, uN * Dr, 1,            // B: m (Dl x uN*Dr)
                 kIn, uN * Dr, Dl);
        __syncthreads();
        // Kronecker accumulate: mat += cm[k,p,Dr] * w[wl,wr,p,d]
        if (cok) {
          const int wbase = ((wl * wrN + wr_) * uN) * dN;
          int kk = 0, dd = 0;
          const int uDr = uN * Dr;
#pragma unroll 1
          for (int row = 0; row < rows; ++row) {
            float acc = mat[row * cols + col];
            const int tbase = kk * uDr + dr_;
#pragma unroll
            for (int p = 0; p < DB; ++p) {
              if (p < uN)
                acc += Tb[tbase + p * Dr] * wB[wbase + p * dN + dd];
            }
            mat[row * cols + col] = acc;
            if (++dd == dN) { dd = 0; ++kk; }
          }
        }
      }
      __syncthreads();
      // mat += 1e-7 * I (reference regularizer)
      const int mn = rows < cols ? rows : cols;
      for (int e = tid; e < mn; e += THREADS) mat[e * cols + e] += 1e-7f;
      __syncthreads();

      if (c == CGRID - 1) {
        // Last site: keep theta untruncated: new site (kIn, dN, cols)
        for (int e = tid; e < rows * cols; e += THREADS)
          wss[c * SITE_F + e] = mat[e];
        if (tid == 0) { dI[c] = kIn; dI[CGRID + c] = cols; }
        __syncthreads();
      } else {
        int kq = CHI;
        if (rows < kq) kq = rows;
        if (cols < kq) kq = cols;

        // Q init = mat[:, :kq]
        for (int e = tid; e < rows * KQ; e += THREADS) {
          const int j = e % KQ, row = e / KQ;
          Qb[e] = (j < kq && j < cols) ? mat[row * cols + j] : 0.f;
        }
        gs_orth(Qb, Tb, Gb, rows, kq);

        // One power iteration: Q <- orth(mat @ (mat^T @ Q))
        gemm_f32(Zb, KQ, mat, 1, cols, Qb, KQ, 1, cols, kq, rows); // Z = mat^T Q
        __syncthreads();
        gemm_f32(Qb, KQ, mat, cols, 1, Zb, KQ, 1, rows, kq, cols); // Q = mat Z
        gs_orth(Qb, Tb, Gb, rows, kq);

        // carry = Q^T @ mat (kq x cols), layout (k, Dl', wl') for next site
        gemm_f32(Zb, cols, Qb, 1, KQ, mat, cols, 1, kq, cols, rows);
        __syncthreads();

        // New boundary site (kIn, dN, kq) = Q reshaped
        for (int e = tid; e < rows * KQ; e += THREADS) {
          const int j = e % KQ, row = e / KQ;
          if (j < kq) wss[c * SITE_F + row * kq + j] = Qb[e];
        }
        if (tid == 0) { dI[c] = kIn; dI[CGRID + c] = kq; }
        __syncthreads();
        kIn = kq;
      }
    }
    __syncthreads();
  }

  // Final contraction: phys==1 on last row; chain v <- v @ site[:,0,:]
  if (tid == 0) vA[0] = 1.0f;
  __syncthreads();
#pragma unroll 1
  for (int c = 0; c < CGRID; ++c) {
    const int Dl = dI[c], Dr = dI[CGRID + c];
    float* src = (c & 1) ? vB2 : vA;
    float* dst = (c & 1) ? vA : vB2;
    if (tid < Dr) {
      float acc = 0.f;
      for (int a = 0; a < Dl; ++a)
        acc += src[a] * wss[c * SITE_F + a * Dr + tid];
      dst[tid] = acc;
    }
    __syncthreads();
  }
  if (tid == 0) out[s] = logf(vA[0]);   // c=9 wrote into vA
}

extern "C" void kernel_launch(void* const* d_in, const int* in_sizes, int n_in,
                              void* d_out, int out_size, void* d_ws, size_t ws_size,
                              hipStream_t stream) {
  (void)in_sizes; (void)out_size; (void)ws_size;
  TensPtrs tp;
  for (int i = 0; i < RGRID * CGRID; ++i)
    tp.t[i] = (const float*)d_in[i];
  const int* x = (const int*)d_in[n_in - 1];
  float* out = (float*)d_out;
  float* ws  = (float*)d_ws;

  const size_t smemBytes =
      (size_t)(MAXR * MAXC + SITE_F + 3 * MAXR * KQ + 1296 + 1296 + 80) * 4
      + 24 * sizeof(int);

  peps_zipup_kernel<<<256, THREADS, smemBytes, stream>>>(tp, x, out, ws);
}